// GraphSparseConvolutionSharingClusterWeights_52536039965272
// MI455X (gfx1250) — compile-verified
//
#include <hip/hip_runtime.h>
#include <hip/hip_bf16.h>

#define NNODES 100000
#define NFEATS 256
#define OUTDIM 64
#define NNZX   1280000
#define NEDGES 1600000
#define NB     (NNODES / 16)   // 6250 row-tiles of 16
#define KEEP_INV (1.0f / 0.9f)
#define XDS    260             // padded K-stride: 260 % 64 == 4 -> conflict-free b64

typedef __attribute__((ext_vector_type(2))) float v2f;
typedef __attribute__((ext_vector_type(8))) float v8f;

// ---------------- threefry2x32 matching jax.random.key(42) -----------------
__device__ __forceinline__ unsigned tf_rotl(unsigned x, int n) {
    return (x << n) | (x >> (32 - n));
}

__device__ unsigned threefry2x32_42(unsigned c0, unsigned c1, bool second) {
    const unsigned ks0 = 0u, ks1 = 42u;
    const unsigned ks2 = ks0 ^ ks1 ^ 0x1BD11BDAu;
    unsigned x0 = c0 + ks0, x1 = c1 + ks1;
#define TFR(r) { x0 += x1; x1 = tf_rotl(x1, r); x1 ^= x0; }
    TFR(13) TFR(15) TFR(26) TFR(6)   x0 += ks1; x1 += ks2 + 1u;
    TFR(17) TFR(29) TFR(16) TFR(24)  x0 += ks2; x1 += ks0 + 2u;
    TFR(13) TFR(15) TFR(26) TFR(6)   x0 += ks0; x1 += ks1 + 3u;
    TFR(17) TFR(29) TFR(16) TFR(24)  x0 += ks1; x1 += ks2 + 4u;
    TFR(13) TFR(15) TFR(26) TFR(6)   x0 += ks2; x1 += ks0 + 5u;
#undef TFR
    return second ? x1 : x0;
}

__device__ __forceinline__ float uniform01_from_key42(int k) {
    const int H = NNZX / 2;
    unsigned bits = (k < H) ? threefry2x32_42((unsigned)k, (unsigned)(k + H), false)
                            : threefry2x32_42((unsigned)(k - H), (unsigned)k, true);
    unsigned f = (bits >> 9) | 0x3f800000u;  // [1,2)
    return __uint_as_float(f) - 1.0f;        // [0,1)
}

// ---------------- Kernel 1: histogram per 16-row tile (shared) -------------
__global__ void k_hist(const int* __restrict__ rows, int n,
                       unsigned* __restrict__ binCount) {
    int k = blockIdx.x * 256 + threadIdx.x;
    if (k < n) atomicAdd(&binCount[rows[k] >> 4], 1u);
}

// ---------------- Kernel 2: exclusive scan over NB bins (single block) -----
__global__ __launch_bounds__(1024) void k_scan(const unsigned* __restrict__ cnt,
                                               unsigned* __restrict__ start,
                                               unsigned* __restrict__ cursor) {
    __shared__ unsigned s[1024];
    int tid = threadIdx.x;
    unsigned carry = 0;
    for (int base = 0; base < NB; base += 1024) {
        int i = base + tid;
        unsigned x = (i < NB) ? cnt[i] : 0u;
        s[tid] = x;
        __syncthreads();
        for (int off = 1; off < 1024; off <<= 1) {
            unsigned t = (tid >= off) ? s[tid - off] : 0u;
            __syncthreads();
            s[tid] += t;
            __syncthreads();
        }
        unsigned excl = s[tid] - x + carry;
        if (i < NB) { start[i] = excl; cursor[i] = excl; }
        unsigned chunkTot = s[1023];
        carry += chunkTot;
        __syncthreads();
    }
    if (tid == 0) start[NB] = carry;
}

// ---------------- Kernel 3a: fused threefry dropout + feature bin scatter --
__global__ void k_scatter_x(const float* __restrict__ vals,
                            const int* __restrict__ rows,
                            const int* __restrict__ cols,
                            unsigned* __restrict__ cursor,
                            float* __restrict__ bVal,
                            unsigned* __restrict__ bIdx) {
    int k = blockIdx.x * 256 + threadIdx.x;
    if (k >= NNZX) return;
    float u    = uniform01_from_key42(k);
    float keep = floorf(0.9f + u);              // exactly 0.0 or 1.0
    float v    = keep * vals[k] * KEEP_INV;
    int r = rows[k], c = cols[k];
    unsigned pos = atomicAdd(&cursor[r >> 4], 1u);
    bVal[pos] = v;
    bIdx[pos] = ((unsigned)(r & 15) << 8) | (unsigned)c;   // [rowmod:4][col:8]
}

// ---------------- Kernel 3b: edge bin scatter ------------------------------
__global__ void k_scatter_e(const float* __restrict__ adj_v,
                            const int* __restrict__ adj_r,
                            const int* __restrict__ adj_c,
                            unsigned* __restrict__ cursor,
                            float* __restrict__ eVal,
                            unsigned* __restrict__ eIdx) {
    int e = blockIdx.x * 256 + threadIdx.x;
    if (e >= NEDGES) return;
    int r = adj_r[e];
    unsigned pos = atomicAdd(&cursor[r >> 4], 1u);
    eVal[pos] = adj_v[e];
    eIdx[pos] = ((unsigned)(r & 15) << 20) | (unsigned)adj_c[e]; // col < 2^17
}

// ---------------- Kernel 4: SpMM1 via f32 WMMA over densified tiles --------
__global__ __launch_bounds__(128) void k_spmm1_wmma(const float* __restrict__ W,
                                                    const unsigned* __restrict__ binStart,
                                                    const float* __restrict__ bVal,
                                                    const unsigned* __restrict__ bIdx,
                                                    float* __restrict__ xw) {
    __shared__ float Xd[16 * XDS];     // densified X tile, [m][k] padded, ~16.6 KB
    __shared__ float Wt[OUTDIM * XDS]; // W transposed,     [n][k] padded, ~66.6 KB
    const int tid = threadIdx.x;
    const int b   = blockIdx.x;

    // zero the X tile
    for (int i = tid; i < 16 * XDS; i += 128) Xd[i] = 0.0f;
    // stage W (row-major [k][n]) transposed into LDS [n][k]
    const float4* W4 = (const float4*)W;
    for (int i = tid; i < (NFEATS * OUTDIM) / 4; i += 128) {
        float4 w = W4[i];
        int k  = i >> 4;           // 16 float4 per K row
        int n4 = (i & 15) << 2;
        Wt[(n4 + 0) * XDS + k] = w.x;
        Wt[(n4 + 1) * XDS + k] = w.y;
        Wt[(n4 + 2) * XDS + k] = w.z;
        Wt[(n4 + 3) * XDS + k] = w.w;
    }
    __syncthreads();

    // densify this tile's nonzeros (duplicates accumulate -> LDS atomics)
    unsigned s0 = binStart[b], s1 = binStart[b + 1];
    for (unsigned i = s0 + tid; i < s1; i += 128) {
        unsigned p = bIdx[i];
        atomicAdd(&Xd[(p >> 8) * XDS + (p & 255u)], bVal[i]);
    }
    __syncthreads();

    // 4 waves; wave w owns output N-tile [16w, 16w+16)
    const int wave = tid >> 5, lane = tid & 31;
    const int half = lane >> 4, m = lane & 15;
    const int n0 = wave * 16;

    // fragment base pointers: contiguous (K, K+1) pairs -> single ds_load_b64
    const float* ap = &Xd[m * XDS + 2 * half];
    const float* bp = &Wt[(n0 + m) * XDS + 2 * half];

    v8f c = {};
#pragma unroll 8
    for (int k0 = 0; k0 < NFEATS; k0 += 4) {
        v2f a  = *(const v2f*)(ap + k0);   // A: K=k0+2h, k0+2h+1 for row m
        v2f bb = *(const v2f*)(bp + k0);   // B: K=k0+2h, k0+2h+1 for col n0+m
        c = __builtin_amdgcn_wmma_f32_16x16x4_f32(
                false, a, false, bb, (short)0, c, false, false);
    }

    // C/D 16x16 f32: VGPR j -> M=j (lanes 0-15) / M=j+8 (lanes 16-31), N=lane%16
    float* outp = xw + (size_t)b * 16 * OUTDIM;
#pragma unroll
    for (int j = 0; j < 8; j++) {
        outp[(j + 8 * half) * OUTDIM + n0 + m] = c[j];
    }
}

// ---------------- Kernel 5: SpMM2 per-tile LDS accumulation + fused ReLU ---
__global__ __launch_bounds__(256) void k_spmm2_tile(const unsigned* __restrict__ eStart,
                                                    const float* __restrict__ eVal,
                                                    const unsigned* __restrict__ eIdx,
                                                    const float* __restrict__ xw,
                                                    float* __restrict__ out) {
    __shared__ float T[16 * OUTDIM];   // 4 KB output tile accumulator
    const int tid = threadIdx.x;
    const int b   = blockIdx.x;

    for (int i = tid; i < 16 * OUTDIM; i += 256) T[i] = 0.0f;
    __syncthreads();

    unsigned s0 = eStart[b], s1 = eStart[b + 1];
    const int l = tid & 15;                       // 16 lanes per edge, 4 cols each
    const float4* xw4 = (const float4*)xw;
    for (unsigned i = s0 + (tid >> 4); i < s1; i += 16) {  // 16 edges in flight
        unsigned p = eIdx[i];
        float    a = eVal[i];
        unsigned c = p & 0xFFFFFu;
        unsigned m = p >> 20;
        float4 x = xw4[(size_t)c * 16 + l];       // coalesced 256B row gather (L2)
        float* t = &T[m * OUTDIM + l * 4];
        atomicAdd(t + 0, a * x.x);                // ds_add_f32, banks 4l..4l+3
        atomicAdd(t + 1, a * x.y);
        atomicAdd(t + 2, a * x.z);
        atomicAdd(t + 3, a * x.w);
    }
    __syncthreads();

    // single streaming write of the tile with fused ReLU
    float4*       o4 = (float4*)(out + (size_t)b * 16 * OUTDIM);
    const float4* t4 = (const float4*)T;
    for (int i = tid; i < (16 * OUTDIM) / 4; i += 256) {
        float4 v = t4[i];
        v.x = fmaxf(v.x, 0.0f); v.y = fmaxf(v.y, 0.0f);
        v.z = fmaxf(v.z, 0.0f); v.w = fmaxf(v.w, 0.0f);
        o4[i] = v;
    }
}

// ---------------------------------------------------------------------------
extern "C" void kernel_launch(void* const* d_in, const int* in_sizes, int n_in,
                              void* d_out, int out_size, void* d_ws, size_t ws_size,
                              hipStream_t stream) {
    const float* feat_values = (const float*)d_in[0];
    const float* kernelW     = (const float*)d_in[1];
    const float* adj_values  = (const float*)d_in[2];
    const int*   feat_rows   = (const int*)d_in[3];
    const int*   feat_cols   = (const int*)d_in[4];
    const int*   adj_rows    = (const int*)d_in[5];
    const int*   adj_cols    = (const int*)d_in[6];
    float*       out         = (float*)d_out;

    // workspace layout (~49 MB)
    char* ws = (char*)d_ws;
    size_t off = 0;
    float*    bVal      = (float*)(ws + off);    off += (size_t)NNZX * 4;
    unsigned* bIdx      = (unsigned*)(ws + off); off += (size_t)NNZX * 4;
    float*    eVal      = (float*)(ws + off);    off += (size_t)NEDGES * 4;
    unsigned* eIdx      = (unsigned*)(ws + off); off += (size_t)NEDGES * 4;
    float*    xw        = (float*)(ws + off);    off += (size_t)NNODES * OUTDIM * 4;
    unsigned* xCount    = (unsigned*)(ws + off); off += (size_t)(NB + 8) * 4;
    unsigned* xStart    = (unsigned*)(ws + off); off += (size_t)(NB + 8) * 4;
    unsigned* xCursor   = (unsigned*)(ws + off); off += (size_t)(NB + 8) * 4;
    unsigned* eCount    = (unsigned*)(ws + off); off += (size_t)(NB + 8) * 4;
    unsigned* eStart    = (unsigned*)(ws + off); off += (size_t)(NB + 8) * 4;
    unsigned* eCursor   = (unsigned*)(ws + off); off += (size_t)(NB + 8) * 4;
    (void)ws_size; (void)in_sizes; (void)n_in; (void)out_size;

    hipMemsetAsync(xCount, 0, (size_t)NB * sizeof(unsigned), stream);
    hipMemsetAsync(eCount, 0, (size_t)NB * sizeof(unsigned), stream);

    // bin features and edges by 16-row tile
    k_hist<<<(NNZX   + 255) / 256, 256, 0, stream>>>(feat_rows, NNZX,   xCount);
    k_hist<<<(NEDGES + 255) / 256, 256, 0, stream>>>(adj_rows,  NEDGES, eCount);
    k_scan<<<1, 1024, 0, stream>>>(xCount, xStart, xCursor);
    k_scan<<<1, 1024, 0, stream>>>(eCount, eStart, eCursor);
    k_scatter_x<<<(NNZX   + 255) / 256, 256, 0, stream>>>(feat_values, feat_rows,
                                                          feat_cols, xCursor, bVal, bIdx);
    k_scatter_e<<<(NEDGES + 255) / 256, 256, 0, stream>>>(adj_values, adj_rows,
                                                          adj_cols, eCursor, eVal, eIdx);

    // SpMM1: dense 16x256 tiles @ W via f32 WMMA -> xw (stays hot in 192MB L2)
    k_spmm1_wmma<<<NB, 128, 0, stream>>>(kernelW, xStart, bVal, bIdx, xw);

    // SpMM2: per-tile LDS accumulation, no global atomics, fused ReLU
    k_spmm2_tile<<<NB, 256, 0, stream>>>(eStart, eVal, eIdx, xw, out);
}